// RegionExtractor_19181323943974
// MI455X (gfx1250) — compile-verified
//
#include <hip/hip_runtime.h>

// ---------------------------------------------------------------------------
// Foveated 128x128 bilinear region extraction, MI455X (gfx1250).
// TDM (tensor_load_to_lds) stages the per-(image,channel) bounding-box tile
// into LDS once per workgroup; threads bilinear-sample from LDS and write
// float4-coalesced output. Memory-bound: ~25 MB traffic -> ~1 us @ 23.3 TB/s.
// ---------------------------------------------------------------------------

typedef __attribute__((ext_vector_type(4))) unsigned int u32x4;
typedef __attribute__((ext_vector_type(8))) int          i32x8;
typedef __attribute__((ext_vector_type(4))) int          i32x4;

#define RS   128          // region size
#define IMH  512
#define IMW  1024
#define TILE_MAX 130      // max bounding-box edge (<=129, padded)

__device__ __forceinline__ float clampf(float v, float lo, float hi) {
    return fminf(fmaxf(v, lo), hi);
}
__device__ __forceinline__ int clampi(int v, int lo, int hi) {
    return v < lo ? lo : (v > hi ? hi : v);
}

extern "C" __global__ __launch_bounds__(256)
void region_extract_tdm(const float* __restrict__ images,
                        const float* __restrict__ gaze,
                        float* __restrict__ out)
{
    // Dynamic LDS: tile lives at LDS byte offset 0 (matches D#.lds_addr=0).
    extern __shared__ float tile[];

    const int blk = blockIdx.x;       // blk = b*3 + c  (channel plane index)
    const int b   = blk / 3;

    // Gaze is uniform per block (scalar loads).
    const float yc = gaze[2 * b + 0] * (float)IMH;
    const float xc = gaze[2 * b + 1] * (float)IMW;

    const float syH = (float)(IMH - 1) / (float)IMH;
    const float sxW = (float)(IMW - 1) / (float)IMW;

    // Bounding box of the sampled footprint. The clipped index sequences are
    // monotone non-decreasing in r, so min/max come from r=0 / r=127.
    const float yA = clampf(yc - 64.0f,          0.0f, (float)(IMH - 1));
    const float yB = clampf(yc - 64.0f + 127.0f, 0.0f, (float)(IMH - 1));
    const float xA = clampf(xc - 64.0f,          0.0f, (float)(IMW - 1));
    const float xB = clampf(xc - 64.0f + 127.0f, 0.0f, (float)(IMW - 1));

    int ry0 = clampi((int)floorf(yA * syH), 0, IMH - 1);
    int ryL = clampi(clampi((int)floorf(yB * syH), 0, IMH - 1) + 1, 0, IMH - 1);
    int rx0 = clampi((int)floorf(xA * sxW), 0, IMW - 1);
    int rxL = clampi(clampi((int)floorf(xB * sxW), 0, IMW - 1) + 1, 0, IMW - 1);

    // Force into SGPRs (values are identical across lanes).
    ry0 = __builtin_amdgcn_readfirstlane(ry0);
    rx0 = __builtin_amdgcn_readfirstlane(rx0);
    const int tileH = __builtin_amdgcn_readfirstlane(ryL - ry0 + 1);  // <= 129
    const int tileW = __builtin_amdgcn_readfirstlane(rxL - rx0 + 1);  // <= 129

    // -------- TDM: DMA the 2D tile [ry0..ryL] x [rx0..rxL] into LDS --------
    const unsigned wave_id =
        __builtin_amdgcn_readfirstlane((int)(threadIdx.x >> 5));
    if (wave_id == 0) {
        unsigned long long gaddr = (unsigned long long)(const void*)
            (images + (size_t)blk * IMH * IMW + (size_t)ry0 * IMW + rx0);
        const unsigned ga_lo = (unsigned)(gaddr & 0xFFFFFFFFull);
        const unsigned ga_hi = (unsigned)(gaddr >> 32);

        // D# group 0 (128b): count=1 | lds_addr | global_addr[56:0] | type=2
        u32x4 g0;
        g0[0] = 1u;                                   // count=1, user mode
        g0[1] = 0u;                                   // lds_addr = 0 (tile[])
        g0[2] = ga_lo;                                // global_addr[31:0]
        g0[3] = (ga_hi & 0x01FFFFFFu) | (2u << 30);   // addr[56:32] | type=2

        // D# group 1 (256b): 2D tensor/tile description, 4-byte elements.
        i32x8 g1;
        g1[0] = (2 << 16);                            // data_size = 4 bytes
        g1[1] = (IMW & 0xFFFF) << 16;                 // tensor_dim0[15:0]
        g1[2] = (IMW >> 16) | ((IMH & 0xFFFF) << 16); // dim0 hi | dim1 lo
        g1[3] = (IMH >> 16) | (tileW << 16);          // dim1 hi | tile_dim0
        g1[4] = tileH;                                // tile_dim1 (tile_dim2=0)
        g1[5] = IMW;                                  // tensor_dim0_stride lo
        g1[6] = 0;                                    // stride hi | dim1 stride
        g1[7] = 0;

        i32x4 gz4 = {0, 0, 0, 0};                     // groups 2/3: 2D tensor
        i32x8 gz8 = {0, 0, 0, 0, 0, 0, 0, 0};         // extra group (clang-23)

        __builtin_amdgcn_tensor_load_to_lds(g0, g1, gz4, gz4, gz8, 0);
        __builtin_amdgcn_s_wait_tensorcnt(0);
    }
    __syncthreads();

    // -------- bilinear sampling from the LDS tile --------
    // Each thread produces 4 consecutive output pixels -> float4 stores,
    // y-side math amortized 4x. Lanes 0..31 cover one full output row.
    const int lane  = threadIdx.x & 31;   // j block: columns lane*4 .. lane*4+3
    const int wrow  = threadIdx.x >> 5;   // starting row (0..7)
    const int jbase = lane * 4;

    for (int i = wrow; i < RS; i += 8) {
        // y-side (once per 4 outputs)
        const float y  = clampf(yc - 64.0f + (float)i, 0.0f, (float)(IMH - 1));
        const float py = y * syH;
        const float yf = floorf(py);
        const float fy = py - yf;
        const int y0i  = clampi((int)yf, 0, IMH - 1);
        const int y1i  = clampi(y0i + 1, 0, IMH - 1);
        const int r0   = (y0i - ry0) * tileW;
        const int r1   = (y1i - ry0) * tileW;

        float4 res;
        float* rp = (float*)&res;
#pragma unroll
        for (int k = 0; k < 4; ++k) {
            const int j    = jbase + k;
            const float x  = clampf(xc - 64.0f + (float)j, 0.0f,
                                    (float)(IMW - 1));
            const float px = x * sxW;
            const float xf = floorf(px);
            const float fx = px - xf;
            const int x0i  = clampi((int)xf, 0, IMW - 1);
            const int x1i  = clampi(x0i + 1, 0, IMW - 1);
            const int x0l  = x0i - rx0;
            const int x1l  = x1i - rx0;

            const float v00 = tile[r0 + x0l];
            const float v01 = tile[r0 + x1l];
            const float v10 = tile[r1 + x0l];
            const float v11 = tile[r1 + x1l];

            const float top = v00 * (1.0f - fx) + v01 * fx;
            const float bot = v10 * (1.0f - fx) + v11 * fx;
            rp[k] = top * (1.0f - fy) + bot * fy;
        }

        *(float4*)(out + (size_t)blk * (RS * RS) + i * RS + jbase) = res;
    }
}

extern "C" void kernel_launch(void* const* d_in, const int* in_sizes, int n_in,
                              void* d_out, int out_size, void* d_ws, size_t ws_size,
                              hipStream_t stream) {
    const float* images = (const float*)d_in[0];   // (64, 3, 512, 1024) f32
    const float* gaze   = (const float*)d_in[1];   // (64, 2) f32
    float* out          = (float*)d_out;           // (64, 3, 128, 128) f32

    const dim3 grid(64 * 3);
    const dim3 block(256);
    const size_t shmem = (size_t)TILE_MAX * TILE_MAX * sizeof(float); // 67.6 KB

    hipLaunchKernelGGL(region_extract_tdm, grid, block, shmem, stream,
                       images, gaze, out);
}